// MeanPooling_6665789243982
// MI455X (gfx1250) — compile-verified
//
#include <hip/hip_runtime.h>
#include <hip/hip_bf16.h>
#include <stdint.h>

typedef float v2f __attribute__((ext_vector_type(2)));
typedef float v4f __attribute__((ext_vector_type(4)));
typedef float v8f __attribute__((ext_vector_type(8)));

namespace {
constexpr int B = 32, E = 256, L = 2048, D = 256;
constexpr int BM = 128, BN = 64, BK = 32;
constexpr int LDA  = BK + 4;    // 36 words: 16B-aligned rows, conflict-free b64 frag reads
constexpr int LDBT = BK + 2;    // 34 words: transposed B tile, 8B-aligned b64 frag reads
constexpr int KITERS = L / BK;  // 64
}

// Low 32 bits of a generic pointer to LDS == wave-relative LDS byte address.
__device__ __forceinline__ uint32_t lds_off(const void* p) {
  return (uint32_t)(uintptr_t)p;
}

// CDNA5 async copy: global -> LDS, per-lane LDS destination, tracked by ASYNCcnt.
__device__ __forceinline__ void async_load_b128(uint32_t lds, const float* g) {
  asm volatile("global_load_async_to_lds_b128 %0, %1, off"
               :: "v"(lds), "v"((unsigned long long)(uintptr_t)g)
               : "memory");
}

__device__ __forceinline__ void wait_asynccnt0() {
#if __has_builtin(__builtin_amdgcn_s_wait_asynccnt)
  __builtin_amdgcn_s_wait_asynccnt(0);
#else
  asm volatile("s_wait_asynccnt 0x0" ::: "memory");
#endif
}

// C[b] (E x D) = emap[b] (E x L) @ doc[b] (L x D), rows scaled by 1/lens[b][e].
__global__ __launch_bounds__(256) void mean_pool_wmma(
    const float* __restrict__ doc,    // [B, L, D]
    const float* __restrict__ emap,   // [B, E, L]
    const float* __restrict__ lens,   // [B, E]
    float* __restrict__ out)          // [B, E, D]
{
  __shared__ float As [2][BM][LDA];   // A tile, row-major (m, k) — filled by async DMA
  __shared__ float Bst[2][BN][LDBT];  // B tile, TRANSPOSED (n, k) — register-staged

  const int tid    = threadIdx.x;
  const int lane   = tid & 31;          // wave32
  const int laneLo = lane & 15;
  const int laneHi = lane >> 4;         // 0/1 half-wave
  const int wave   = tid >> 5;          // 0..7
  const int wm     = wave >> 1;         // 0..3 : 32-row strip
  const int wn     = wave & 1;          // 0..1 : 32-col strip

  const int bidx = blockIdx.x;          // 32 * (E/BM=2) * (D/BN=4) = 256
  const int b    = bidx >> 3;
  const int tm   = (bidx >> 2) & 1;
  const int tn   = bidx & 3;

  const float* Ag = emap + ((size_t)b * E + (size_t)tm * BM) * L;
  const float* Bg = doc  + (size_t)b * L * D + (size_t)tn * BN;

  // A staging: 128x32 f32 = 16KB -> 4 async b128 per thread (32 rows/pass).
  const int arow = tid >> 3;            // 0..31
  const int acol = (tid & 7) << 2;      // 0,4,...,28
  // B staging: 32x64 f32 -> 2 float4 global loads per thread, transposed scatter.
  const int brow = tid >> 4;            // 0..15 (k rows brow, brow+16)
  const int bcol = (tid & 15) << 2;     // 0,4,...,60 (d columns)

  const v8f vzero = {0.f, 0.f, 0.f, 0.f, 0.f, 0.f, 0.f, 0.f};
  v8f acc[2][2];
  acc[0][0] = vzero; acc[0][1] = vzero; acc[1][0] = vzero; acc[1][1] = vzero;

  v4f rb[2];

  // ---- Prologue: stage K-tile 0 into buffer 0 ----
#pragma unroll
  for (int i = 0; i < 4; ++i)
    async_load_b128(lds_off(&As[0][arow + i * 32][acol]),
                    Ag + (size_t)(arow + i * 32) * L + acol);
#pragma unroll
  for (int i = 0; i < 2; ++i)
    rb[i] = *(const v4f*)(Bg + (size_t)(brow + i * 16) * D + bcol);
#pragma unroll
  for (int i = 0; i < 2; ++i)
#pragma unroll
    for (int j = 0; j < 4; ++j)
      Bst[0][bcol + j][brow + i * 16] = rb[i][j];
  wait_asynccnt0();
  __syncthreads();

  // ---- Main loop: double-buffered, one barrier per K-iteration ----
  for (int t = 0; t < KITERS; ++t) {
    const int cb = t & 1;
    const int nb = cb ^ 1;

    if (t + 1 < KITERS) {
      const int k0 = (t + 1) * BK;
      // Next A tile: async DMA straight into the other LDS buffer.
#pragma unroll
      for (int i = 0; i < 4; ++i)
        async_load_b128(lds_off(&As[nb][arow + i * 32][acol]),
                        Ag + (size_t)(arow + i * 32) * L + k0 + acol);
      // Next B tile: global -> registers (latency hidden under the WMMAs below).
#pragma unroll
      for (int i = 0; i < 2; ++i)
        rb[i] = *(const v4f*)(Bg + (size_t)(k0 + brow + i * 16) * D + bcol);
    }

    // 8 WMMA K-steps of 4 over the current BK=32 tile.
#pragma unroll
    for (int ks = 0; ks < BK / 4; ++ks) {
      const int kf = ks * 4 + 2 * laneHi;  // lanes 0-15: K,K+1 ; lanes 16-31: K+2,K+3
      v2f a0 = *(const v2f*)&As [cb][wm * 32 + laneLo][kf];
      v2f a1 = *(const v2f*)&As [cb][wm * 32 + 16 + laneLo][kf];
      v2f b0 = *(const v2f*)&Bst[cb][wn * 32 + laneLo][kf];
      v2f b1 = *(const v2f*)&Bst[cb][wn * 32 + 16 + laneLo][kf];
      acc[0][0] = __builtin_amdgcn_wmma_f32_16x16x4_f32(false, a0, false, b0,
                                                        (short)0, acc[0][0], false, false);
      acc[0][1] = __builtin_amdgcn_wmma_f32_16x16x4_f32(false, a0, false, b1,
                                                        (short)0, acc[0][1], false, false);
      acc[1][0] = __builtin_amdgcn_wmma_f32_16x16x4_f32(false, a1, false, b0,
                                                        (short)0, acc[1][0], false, false);
      acc[1][1] = __builtin_amdgcn_wmma_f32_16x16x4_f32(false, a1, false, b1,
                                                        (short)0, acc[1][1], false, false);
    }

    if (t + 1 < KITERS) {
      // Transposed scatter of the next B tile into the other buffer
      // (safe pre-barrier: nb's last readers finished at the previous barrier).
#pragma unroll
      for (int i = 0; i < 2; ++i)
#pragma unroll
        for (int j = 0; j < 4; ++j)
          Bst[nb][bcol + j][brow + i * 16] = rb[i][j];
      wait_asynccnt0();   // next A tile fully landed in LDS
    }
    __syncthreads();
  }

  // ---- Epilogue: scale by 1/lens and store (coalesced 64B per half-wave) ----
  const float* lenp = lens + (size_t)b * E + (size_t)tm * BM;
  float* Og = out + ((size_t)b * E + (size_t)tm * BM) * D + (size_t)tn * BN;
#pragma unroll
  for (int mt = 0; mt < 2; ++mt) {
#pragma unroll
    for (int v = 0; v < 8; ++v) {
      const int row = wm * 32 + mt * 16 + v + 8 * laneHi;
      const float invl = 1.0f / lenp[row];
#pragma unroll
      for (int nt = 0; nt < 2; ++nt) {
        const int col = wn * 32 + nt * 16 + laneLo;
        Og[(size_t)row * D + col] = acc[mt][nt][v] * invl;
      }
    }
  }
}

extern "C" void kernel_launch(void* const* d_in, const int* in_sizes, int n_in,
                              void* d_out, int out_size, void* d_ws, size_t ws_size,
                              hipStream_t stream) {
  (void)in_sizes; (void)n_in; (void)d_ws; (void)ws_size; (void)out_size;
  const float* doc  = (const float*)d_in[0];   // doc_state      [B, L, D]
  const float* emap = (const float*)d_in[1];   // entity_mapping [B, E, L]
  const float* lens = (const float*)d_in[2];   // entity_lens    [B, E]
  float* out = (float*)d_out;                  // [B, E, D]

  dim3 grid(B * (E / BM) * (D / BN));  // 256 blocks
  mean_pool_wmma<<<grid, 256, 0, stream>>>(doc, emap, lens, out);
}